// ROIExtractBaseline_14070312861892
// MI455X (gfx1250) — compile-verified
//
#include <hip/hip_runtime.h>
#include <stdint.h>

// ROI crop + bilinear resize (ROIAlign-style), matches reference semantics:
//   src = (o + 0.5) * box / 224 - 0.5, clamped to crop interior, 4-tap bilinear.
// Fixed problem geometry from the reference: C=3, H=1080, W=1920, OUT=224,
// box dims in [64, 320). N derived from in_sizes[1]/4.

#define ROI_OUT 224
#define ROI_C   3
#define ROI_H   1080
#define ROI_W   1920
#define LDSW    336   // padded LDS row stride (floats); >= 324 needed
#define NSROW   6     // 2 source rows x 3 channels

// Exact pointee types for the async-copy builtins (from clang diagnostics:
// param 0 is "__attribute__((vector_size(16))) int" in AS(1)).
typedef int v4i_t __attribute__((vector_size(16)));
typedef __attribute__((address_space(1))) v4i_t* g_v4i_ptr;
typedef __attribute__((address_space(3))) v4i_t* l_v4i_ptr;
typedef __attribute__((address_space(1))) int*   g_i32_ptr;
typedef __attribute__((address_space(3))) int*   l_i32_ptr;

// ---- gfx1250 async global->LDS path (probe-guarded) ----
#if defined(__has_builtin)
#if __has_builtin(__builtin_amdgcn_global_load_async_to_lds_b128)
#define USE_ASYNC_B128 1
#elif __has_builtin(__builtin_amdgcn_global_load_async_to_lds_b32)
#define USE_ASYNC_B32 1
#endif
#endif

#if defined(USE_ASYNC_B128) || defined(USE_ASYNC_B32)
#define USE_ASYNC 1
#endif

__device__ __forceinline__ void wait_async_zero() {
#if defined(USE_ASYNC)
#if __has_builtin(__builtin_amdgcn_s_wait_asynccnt)
    __builtin_amdgcn_s_wait_asynccnt(0);
#else
    asm volatile("s_wait_asynccnt 0" ::: "memory");
#endif
#endif
}

__global__ __launch_bounds__(256)
void roi_crop_resize_kernel(const float* __restrict__ feat,
                            const int*  __restrict__ det,
                            float* __restrict__ out)
{
    __shared__ float smem[NSROW * LDSW];

    const int oy  = blockIdx.x;      // output row 0..223
    const int n   = blockIdx.y;      // ROI index
    const int tid = threadIdx.x;

    const int x1 = det[n * 4 + 0];
    const int y1 = det[n * 4 + 1];
    const int x2 = det[n * 4 + 2];
    const int y2 = det[n * 4 + 3];
    const float bw = (float)(x2 - x1);
    const float bh = (float)(y2 - y1);

    // Vertical source coordinate (align_corners=False, clamped to crop).
    float ys = (oy + 0.5f) * bh * (1.0f / ROI_OUT) - 0.5f;
    ys = fminf(fmaxf(ys, 0.0f), bh - 1.0f) + (float)y1;
    const float y0f = floorf(ys);
    const float wy  = ys - y0f;
    int y0 = (int)y0f; y0 = min(max(y0, 0), ROI_H - 1);
    const int y1r = min(y0 + 1, ROI_H - 1);

    // Staged column window [calign, calign+ccount), 16B-aligned, covers [x1, x2].
    const int shift  = x1 & 3;
    const int calign = x1 - shift;
    int ccount = (x2 - x1) + 1 + shift;   // columns needed incl. alignment shift
    ccount = (ccount + 3) & ~3;           // round up to 4-float vectors
    ccount = min(ccount, ROI_W - calign); // W%4==0 keeps this a multiple of 4
    const int nvec = ccount >> 2;

    // ---- stage 6 source rows into LDS (async DMA path on gfx1250) ----
    for (int r = 0; r < NSROW; ++r) {
        const int c  = r >> 1;
        const int yy = (r & 1) ? y1r : y0;
        const float* gsrc = feat + (((size_t)n * ROI_C + c) * ROI_H + yy) * ROI_W + calign;
        float* lrow = &smem[r * LDSW];
#if defined(USE_ASYNC_B128)
        for (int v = tid; v < nvec; v += 256) {
            __builtin_amdgcn_global_load_async_to_lds_b128(
                (g_v4i_ptr)(gsrc + 4 * v),
                (l_v4i_ptr)(lrow + 4 * v),
                0, 0);
        }
#elif defined(USE_ASYNC_B32)
        for (int col = tid; col < ccount; col += 256) {
            __builtin_amdgcn_global_load_async_to_lds_b32(
                (g_i32_ptr)(gsrc + col),
                (l_i32_ptr)(lrow + col),
                0, 0);
        }
#else
        for (int col = tid; col < ccount; col += 256)
            lrow[col] = gsrc[col];
#endif
        (void)nvec;
    }
    wait_async_zero();   // each wave drains its own ASYNCcnt, then barrier
    __syncthreads();

    // ---- bilinear resample from LDS: 3 channels x 224 pixels = 672 outputs ----
    const float wyc = 1.0f - wy;
    for (int o = tid; o < ROI_C * ROI_OUT; o += 256) {
        const int c  = o / ROI_OUT;
        const int ox = o - c * ROI_OUT;
        float xs = (ox + 0.5f) * bw * (1.0f / ROI_OUT) - 0.5f;
        xs = fminf(fmaxf(xs, 0.0f), bw - 1.0f);
        const float x0f = floorf(xs);
        const float wx  = xs - x0f;
        const int ix0 = (int)x0f + shift;   // local column in staged window
        const int ix1 = ix0 + 1;            // in-bounds: ccount >= shift+bw+1
        const float* r0 = &smem[(c * 2 + 0) * LDSW];
        const float* r1 = &smem[(c * 2 + 1) * LDSW];
        const float v00 = r0[ix0], v01 = r0[ix1];
        const float v10 = r1[ix0], v11 = r1[ix1];
        const float top = v00 * (1.0f - wx) + v01 * wx;
        const float bot = v10 * (1.0f - wx) + v11 * wx;
        out[(((size_t)n * ROI_C + c) * ROI_OUT + oy) * ROI_OUT + ox] = top * wyc + bot * wy;
    }
}

extern "C" void kernel_launch(void* const* d_in, const int* in_sizes, int n_in,
                              void* d_out, int out_size, void* d_ws, size_t ws_size,
                              hipStream_t stream) {
    (void)n_in; (void)out_size; (void)d_ws; (void)ws_size;
    const float* feat = (const float*)d_in[0];
    const int*   det  = (const int*)d_in[1];
    float*       out  = (float*)d_out;
    const int N = in_sizes[1] / 4;          // detections are [N,4] int32
    dim3 grid(ROI_OUT, N);                  // one block per (output row, ROI)
    roi_crop_resize_kernel<<<grid, dim3(256), 0, stream>>>(feat, det, out);
}